// RGNNLayer_80221399155533
// MI455X (gfx1250) — compile-verified
//
#include <hip/hip_runtime.h>
#include <hip/hip_bf16.h>

typedef __attribute__((ext_vector_type(2))) float v2f;
typedef __attribute__((ext_vector_type(8))) float v8f;

#define D_FEAT 128
#define UNITS  128

// ---------------------------------------------------------------------------
// GEMM: Y[nRows x 128] = X[nRows x 128] @ W[128 x 128], fp32 WMMA 16x16x4.
// Block = 256 threads = 8 wave32s, covering 64 rows x 128 cols.
// Each wave owns a 16-column stripe. Its entire B operand (128x16 stripe of W)
// is 64 floats/lane -> preloaded once into 32 v2f register fragments and
// reused across all 4 row-tiles, so the inner loop is pure LDS-A + WMMA.
// X rows staged in LDS with stride 130 (even -> float2 aligned; breaks the
// 16-way bank conflict of row-strided A-fragment reads).
// ---------------------------------------------------------------------------
__global__ void rgcn_gemm_wmma(const float* __restrict__ X,
                               const float* __restrict__ W,
                               float* __restrict__ Y,
                               int nRows)
{
    __shared__ float sX[64 * 130];   // 33,280 B

    const int tid      = threadIdx.x;        // 0..255
    const int wave     = tid >> 5;           // 0..7
    const int lane     = tid & 31;
    const int rowBlock = blockIdx.x * 64;

    const int colBase = wave * 16;     // this wave's 16 output columns
    const int half    = lane >> 4;     // 0: K pair {k,k+1} / M,N rows 0..7 ; 1: {k+2,k+3} / 8..15
    const int lm      = lane & 15;

    // Cooperative load: 64 rows x 128 cols = 2048 float4, 8 per thread.
    for (int i = tid; i < 64 * 32; i += 256) {
        const int r   = i >> 5;      // row within block
        const int c4  = i & 31;      // float4 index within row
        const int row = rowBlock + r;
        float4 v = make_float4(0.f, 0.f, 0.f, 0.f);
        if (row < nRows)
            v = *(const float4*)(X + (size_t)row * D_FEAT + c4 * 4);
        float* dst = &sX[r * 130 + c4 * 4];
        dst[0] = v.x; dst[1] = v.y; dst[2] = v.z; dst[3] = v.w;
    }

    // Preload this wave's full B stripe into registers (32 k-steps x v2f).
    // B fragment (4x16 layout): lane holds W[kk][N=lm], W[kk+1][N=lm].
    v2f bfrag[32];
    #pragma unroll
    for (int ks = 0; ks < 32; ++ks) {
        const int kk = ks * 4 + 2 * half;
        bfrag[ks].x = W[kk * UNITS + colBase + lm];
        bfrag[ks].y = W[(kk + 1) * UNITS + colBase + lm];
    }

    __syncthreads();

    for (int rt = 0; rt < 4; ++rt) {
        const int rowBase = rowBlock + rt * 16;   // uniform across block
        if (rowBase >= nRows) break;              // uniform -> EXEC stays all-ones

        v8f acc = {};
        #pragma unroll
        for (int ks = 0; ks < 32; ++ks) {
            const int kk = ks * 4 + 2 * half;
            // A fragment (ISA 32-bit A 16x4 layout): lane holds X[M=lm][kk], X[M=lm][kk+1]
            v2f a;
            {
                const float2 av = *(const float2*)&sX[(rt * 16 + lm) * 130 + kk];
                a.x = av.x; a.y = av.y;
            }
            acc = __builtin_amdgcn_wmma_f32_16x16x4_f32(
                false, a, false, bfrag[ks], (short)0, acc, false, false);
        }

        // C/D layout: VGPR r -> row (half*8 + r), col lm
        #pragma unroll
        for (int r = 0; r < 8; ++r) {
            Y[(size_t)(rowBase + half * 8 + r) * UNITS + colBase + lm] = acc[r];
        }
    }
}

// ---------------------------------------------------------------------------
// Undirected scatter-add of T rows into out: one wave32 per edge, both
// directions. Two coalesced 512B float4 gathers + 8 f32 atomics per wave.
// ---------------------------------------------------------------------------
__global__ void rgcn_scatter_add(const float* __restrict__ T,
                                 const int* __restrict__ ea,
                                 const int* __restrict__ eb,
                                 float* __restrict__ out,
                                 int nEdges)
{
    const int e    = (int)((blockIdx.x * (unsigned)blockDim.x + threadIdx.x) >> 5);
    const int lane = threadIdx.x & 31;
    if (e >= nEdges) return;

    const int a = ea[e];
    const int b = eb[e];

    const float4 va = *(const float4*)(T + (size_t)a * UNITS + lane * 4);
    const float4 vb = *(const float4*)(T + (size_t)b * UNITS + lane * 4);

    float* pa = out + (size_t)a * UNITS + lane * 4;
    float* pb = out + (size_t)b * UNITS + lane * 4;

    // out[a] += T[b]
    atomicAdd(pa + 0, vb.x);
    atomicAdd(pa + 1, vb.y);
    atomicAdd(pa + 2, vb.z);
    atomicAdd(pa + 3, vb.w);
    // out[b] += T[a]
    atomicAdd(pb + 0, va.x);
    atomicAdd(pb + 1, va.y);
    atomicAdd(pb + 2, va.z);
    atomicAdd(pb + 3, va.w);
}

// ---------------------------------------------------------------------------
// out = out * norms[row] + bias[col], float4-vectorized.
// ---------------------------------------------------------------------------
__global__ void rgcn_finalize(float* __restrict__ out,
                              const float* __restrict__ norms,
                              const float* __restrict__ bias,
                              int nRows)
{
    const int idx = blockIdx.x * blockDim.x + threadIdx.x;  // float4 index
    const int n4  = nRows * (UNITS / 4);
    if (idx >= n4) return;
    const int row = idx >> 5;          // / (UNITS/4)
    const int c4  = idx & 31;

    float4 v = *(float4*)(out + (size_t)idx * 4);
    const float s = norms[row];
    const float4 bb = *(const float4*)(bias + c4 * 4);
    v.x = v.x * s + bb.x;
    v.y = v.y * s + bb.y;
    v.z = v.z * s + bb.z;
    v.w = v.w * s + bb.w;
    *(float4*)(out + (size_t)idx * 4) = v;
}

extern "C" void kernel_launch(void* const* d_in, const int* in_sizes, int n_in,
                              void* d_out, int out_size, void* d_ws, size_t ws_size,
                              hipStream_t stream) {
    const float* X     = (const float*)d_in[0];   // nNodes x 128
    const int*   ref_a = (const int*)  d_in[1];   // 3 x nEdges
    const int*   ref_b = (const int*)  d_in[2];   // 3 x nEdges
    const float* norms = (const float*)d_in[3];   // nNodes
    const float* W0    = (const float*)d_in[4];   // 128 x 128
    const float* W     = (const float*)d_in[5];   // 3 x 128 x 128
    const float* bias  = (const float*)d_in[6];   // 128

    float* out = (float*)d_out;
    float* T   = (float*)d_ws;                    // nNodes x 128 scratch (51.2 MB)

    const int nNodes = in_sizes[3];               // 100000
    const int nRel   = 3;
    const int nEdges = in_sizes[1] / nRel;        // 500000

    const int gemmGrid = (nNodes + 63) / 64;

    // out = X @ W0   (fully overwrites d_out)
    rgcn_gemm_wmma<<<gemmGrid, 256, 0, stream>>>(X, W0, out, nNodes);

    for (int r = 0; r < nRel; ++r) {
        // T = X @ W[r]
        rgcn_gemm_wmma<<<gemmGrid, 256, 0, stream>>>(
            X, W + (size_t)r * D_FEAT * UNITS, T, nNodes);
        // out[a] += T[b]; out[b] += T[a]  (one wave per edge)
        const int blocks = (nEdges * 32 + 255) / 256;
        rgcn_scatter_add<<<blocks, 256, 0, stream>>>(
            T, ref_a + (size_t)r * nEdges, ref_b + (size_t)r * nEdges, out, nEdges);
    }

    // out = out * norms + bias
    const int n4 = nNodes * (UNITS / 4);
    rgcn_finalize<<<(n4 + 255) / 256, 256, 0, stream>>>(out, norms, bias, nNodes);
}